// GATv2ConvThr_76690936037711
// MI455X (gfx1250) — compile-verified
//
#include <hip/hip_runtime.h>

#define F 128
#define HEADS 8
#define NEG_SLOPE 0.2f
#define KEY_NEG_INF 0x007FFFFFu  // order-preserving key of -inf

typedef __attribute__((ext_vector_type(2))) float v2f;
typedef __attribute__((ext_vector_type(4))) float v4f;
typedef __attribute__((ext_vector_type(8))) float v8f;

// Monotonic float<->uint mapping so atomicMax(uint) == float max (exact, deterministic)
__device__ __forceinline__ unsigned f32_key(float f) {
    unsigned b = __float_as_uint(f);
    return (b & 0x80000000u) ? ~b : (b | 0x80000000u);
}
__device__ __forceinline__ float key_f32(unsigned k) {
    unsigned b = (k & 0x80000000u) ? (k & 0x7FFFFFFFu) : ~k;
    return __uint_as_float(b);
}

// ---------------------------------------------------------------------------
// Projection GEMM: out[n,128] = x[n,128] @ W^T (W is [128,128]) in full f32
// via V_WMMA_F32_16X16X4_F32. One wave -> 16x128 output strip.
// A 16x4 layout: lane holds row M=lane&15; VGPR v holds K = k0 + (lane>>4)*2 + v
// B 4x16 layout: lane holds col N=lane&15 => per-lane loads row N of W (B=W^T)
// C/D: lane&15 = N; VGPR r holds M = (lane>>4)*8 + r
// ---------------------------------------------------------------------------
__global__ __launch_bounds__(32) void gat_proj_wmma(
    const float* __restrict__ x, const float* __restrict__ Wl,
    const float* __restrict__ Wr, float* __restrict__ xl,
    float* __restrict__ xr, int n) {
    const int lane  = threadIdx.x & 31;
    const int row   = lane & 15;
    const int half  = lane >> 4;
    const int mbase = blockIdx.x * 16;

    const float* W   = (blockIdx.y == 0) ? Wl : Wr;
    float*       dst = (blockIdx.y == 0) ? xl : xr;

    int arow = mbase + row;
    if (arow >= n) arow = n - 1;                       // clamp (n==50000 is 16-aligned anyway)
    const float* xrow = x + (size_t)arow * F + half * 2;
    const float* wrow = W + (size_t)row  * F + half * 2;

    v8f acc[8];
#pragma unroll
    for (int t = 0; t < 8; ++t) acc[t] = (v8f){0.f, 0.f, 0.f, 0.f, 0.f, 0.f, 0.f, 0.f};

    for (int k0 = 0; k0 < F; k0 += 4) {
        v2f a = *(const v2f*)(xrow + k0);
#pragma unroll
        for (int t = 0; t < 8; ++t) {
            v2f b = *(const v2f*)(wrow + (size_t)t * 16 * F + k0);
            acc[t] = __builtin_amdgcn_wmma_f32_16x16x4_f32(
                false, a, false, b, (short)0, acc[t], false, false);
        }
    }

#pragma unroll
    for (int t = 0; t < 8; ++t) {
#pragma unroll
        for (int r = 0; r < 8; ++r) {
            const int mrow = mbase + half * 8 + r;
            if (mrow < n) dst[(size_t)mrow * F + t * 16 + row] = acc[t][r];
        }
    }
}

// ---------------------------------------------------------------------------
// Init: out = bias (broadcast), amax = key(-inf), denom = 0
// ---------------------------------------------------------------------------
__global__ void gat_init(float* __restrict__ out, const float* __restrict__ bias,
                         unsigned* __restrict__ amax, float* __restrict__ denom, int n) {
    const int i = blockIdx.x * blockDim.x + threadIdx.x;
    if (i < n * F) out[i] = bias[i & (F - 1)];
    if (i < n * HEADS) { amax[i] = KEY_NEG_INF; denom[i] = 0.f; }
}

// ---------------------------------------------------------------------------
// Per-edge attention logits: one wave per edge. lane owns channels [4*lane,4*lane+4)
// (4 lanes per head). alpha[e,h] = sum_c att[h,c]*lrelu(x_i + x_j); segment max.
// ---------------------------------------------------------------------------
__global__ void gat_edge_alpha(const float* __restrict__ xl, const float* __restrict__ xr,
                               const int* __restrict__ src, const int* __restrict__ dst,
                               const float* __restrict__ att, float* __restrict__ alpha,
                               unsigned* __restrict__ amax, int m) {
    const int e    = (blockIdx.x * blockDim.x + threadIdx.x) >> 5;
    const int lane = threadIdx.x & 31;
    if (e >= m) return;
    const int s = src[e], d = dst[e];
    const int c0 = lane * 4;

    v4f xj = *(const v4f*)(xl  + (size_t)s * F + c0);
    v4f xi = *(const v4f*)(xr  + (size_t)d * F + c0);
    v4f av = *(const v4f*)(att + c0);               // flat att idx == channel idx

    float p = 0.f;
#pragma unroll
    for (int j = 0; j < 4; ++j) {
        float v = xi[j] + xj[j];
        v = (v > 0.f) ? v : NEG_SLOPE * v;
        p += av[j] * v;
    }
    p += __shfl_xor(p, 1, 32);
    p += __shfl_xor(p, 2, 32);                      // 4-lane (per-head) reduction

    if ((lane & 3) == 0) {
        const int h = lane >> 2;
        alpha[(size_t)e * HEADS + h] = p;
        atomicMax(amax + (size_t)d * HEADS + h, f32_key(p));
    }
}

// ---------------------------------------------------------------------------
// ea = exp(alpha - amax[dst]); denom += ea. One thread per (edge, head).
// ---------------------------------------------------------------------------
__global__ void gat_edge_exp(const int* __restrict__ dst, float* __restrict__ alpha,
                             const unsigned* __restrict__ amax,
                             float* __restrict__ denom, int m) {
    const int i = blockIdx.x * blockDim.x + threadIdx.x;
    if (i >= m * HEADS) return;
    const int e = i >> 3, h = i & 7;
    const int d = dst[e];
    const float mx = key_f32(amax[(size_t)d * HEADS + h]);
    const float ea = __expf(alpha[i] - mx);
    alpha[i] = ea;                                   // alpha buffer now holds ea
    atomicAdd(denom + (size_t)d * HEADS + h, ea);
}

// ---------------------------------------------------------------------------
// Scatter-aggregate: out[dst] += (ea/denom) * x_j. One wave per edge.
// ---------------------------------------------------------------------------
__global__ void gat_edge_aggr(const float* __restrict__ xl, const int* __restrict__ src,
                              const int* __restrict__ dst, const float* __restrict__ alpha,
                              const float* __restrict__ denom, float* __restrict__ out, int m) {
    const int e    = (blockIdx.x * blockDim.x + threadIdx.x) >> 5;
    const int lane = threadIdx.x & 31;
    if (e >= m) return;
    const int s = src[e], d = dst[e];
    const int c0 = lane * 4;
    const int h  = lane >> 2;

    const float w = alpha[(size_t)e * HEADS + h] /
                    (denom[(size_t)d * HEADS + h] + 1e-16f);
    v4f xj = *(const v4f*)(xl + (size_t)s * F + c0);
    float* op = out + (size_t)d * F + c0;
    atomicAdd(op + 0, w * xj[0]);
    atomicAdd(op + 1, w * xj[1]);
    atomicAdd(op + 2, w * xj[2]);
    atomicAdd(op + 3, w * xj[3]);
}

// ---------------------------------------------------------------------------
extern "C" void kernel_launch(void* const* d_in, const int* in_sizes, int n_in,
                              void* d_out, int out_size, void* d_ws, size_t ws_size,
                              hipStream_t stream) {
    const float* x    = (const float*)d_in[0];
    const int*   ei   = (const int*)  d_in[1];
    const float* Wl   = (const float*)d_in[2];
    const float* Wr   = (const float*)d_in[3];
    const float* att  = (const float*)d_in[4];
    const float* bias = (const float*)d_in[5];

    const int n = in_sizes[0] / F;      // 50000
    const int m = in_sizes[1] / 2;      // 800000
    const int* src = ei;                // edge_index[0]
    const int* dst = ei + m;            // edge_index[1]

    // workspace layout (f32 unless noted)
    float*    xl    = (float*)d_ws;                       // n*F
    float*    xr    = xl + (size_t)n * F;                 // n*F
    float*    alpha = xr + (size_t)n * F;                 // m*H (logits, then ea)
    unsigned* amax  = (unsigned*)(alpha + (size_t)m * HEADS); // n*H (uint keys)
    float*    denom = (float*)(amax + (size_t)n * HEADS);     // n*H
    float*    out   = (float*)d_out;

    const int initN = n * F;
    gat_init<<<(initN + 255) / 256, 256, 0, stream>>>(out, bias, amax, denom, n);

    dim3 gproj((n + 15) / 16, 2);
    gat_proj_wmma<<<gproj, 32, 0, stream>>>(x, Wl, Wr, xl, xr, n);

    const int edgeBlocks = (m * 32 + 255) / 256;   // one wave32 per edge
    gat_edge_alpha<<<edgeBlocks, 256, 0, stream>>>(xl, xr, src, dst, att, alpha, amax, m);
    gat_edge_exp<<<(m * HEADS + 255) / 256, 256, 0, stream>>>(dst, alpha, amax, denom, m);
    gat_edge_aggr<<<edgeBlocks, 256, 0, stream>>>(xl, src, dst, alpha, denom, out, m);
}